// RNNModel_47270410060169
// MI455X (gfx1250) — compile-verified
//
#include <hip/hip_runtime.h>

typedef float v2f __attribute__((ext_vector_type(2)));
typedef float v8f __attribute__((ext_vector_type(8)));

#define B_SZ 4096
#define T_SZ 512
#define DIN  8
#define H_SZ 50
#define NT   4    // N tiles of 16 covering padded H=64
#define KC   13   // K chunks of 4 covering k=0..51 (>=50 zero-padded)
#define RS   68   // LDS row stride in floats (padded vs 64: bank-conflict free)

// gfx1250 has a native v_tanh_f32 TRANS op; use the builtin if this clang
// declares it, else fall back to ocml tanhf.
__device__ __forceinline__ float fast_tanh(float v) {
#if __has_builtin(__builtin_amdgcn_tanhf)
    return __builtin_amdgcn_tanhf(v);
#elif __has_builtin(__builtin_amdgcn_tanh_f32)
    return __builtin_amdgcn_tanh_f32(v);
#else
    return tanhf(v);
#endif
}

__global__ __launch_bounds__(32)
void rnn_scan_kernel(const float* __restrict__ x,
                     const float* __restrict__ W_ih,
                     const float* __restrict__ W_hh,
                     const float* __restrict__ b_ih,
                     const float* __restrict__ b_hh,
                     const float* __restrict__ fc_W,
                     const float* __restrict__ fc_b,
                     float* __restrict__ out)
{
    __shared__ float hl[16 * RS];   // one wave per block: private h tile [16 rows][64 cols(+pad)]

    const int lane = threadIdx.x & 31;
    const int lo   = lane & 15;     // column (N) / batch-row (A M) index
    const int hi   = lane >> 4;     // half-wave selector
    const int b0   = blockIdx.x * 16;

    // ---- B-layout weight fragments: VGPR v <-> row k = 4c + v + 2*hi, col j = 16n + lo
    v2f Bih[NT][2];
    #pragma unroll
    for (int n = 0; n < NT; ++n) {
        const int j = 16 * n + lo;
        #pragma unroll
        for (int c = 0; c < 2; ++c) {
            const int k0 = 4 * c + 2 * hi;
            Bih[n][c].x = (j < H_SZ) ? W_ih[j * DIN + k0]     : 0.0f;
            Bih[n][c].y = (j < H_SZ) ? W_ih[j * DIN + k0 + 1] : 0.0f;
        }
    }
    v2f Bhh[NT][KC];
    #pragma unroll
    for (int n = 0; n < NT; ++n) {
        const int j = 16 * n + lo;
        #pragma unroll
        for (int c = 0; c < KC; ++c) {
            const int k0 = 4 * c + 2 * hi;
            Bhh[n][c].x = (j < H_SZ && k0     < H_SZ) ? W_hh[j * H_SZ + k0]     : 0.0f;
            Bhh[n][c].y = (j < H_SZ && k0 + 1 < H_SZ) ? W_hh[j * H_SZ + k0 + 1] : 0.0f;
        }
    }
    // bias (depends on N=j only, same across all 8 C VGPRs)
    float cinit[NT];
    #pragma unroll
    for (int n = 0; n < NT; ++n) {
        const int j = 16 * n + lo;
        cinit[n] = (j < H_SZ) ? (b_ih[j] + b_hh[j]) : 0.0f;
    }

    // x_t A-fragment loader: lane holds x[b0+lo][t][4c+2*hi .. +1]
    const float* xrow = x + (size_t)(b0 + lo) * T_SZ * DIN;
    auto ldx = [&](int t, int c) -> v2f {
        return *(const v2f*)(xrow + (size_t)t * DIN + 4 * c + 2 * hi);
    };

    v2f ha[KC];                       // h A-fragments (h0 = 0)
    #pragma unroll
    for (int c = 0; c < KC; ++c) { ha[c].x = 0.0f; ha[c].y = 0.0f; }

    v2f xa0 = ldx(0, 0), xa1 = ldx(0, 1);

    for (int t = 0; t < T_SZ; ++t) {
        const int tn = (t + 1 < T_SZ) ? t + 1 : t;       // software prefetch next step
        v2f nxa0 = ldx(tn, 0), nxa1 = ldx(tn, 1);

        v8f acc[NT];
        #pragma unroll
        for (int n = 0; n < NT; ++n)
            #pragma unroll
            for (int r = 0; r < 8; ++r) acc[n][r] = cinit[n];

        // xW_ih^T (K=8)
        #pragma unroll
        for (int n = 0; n < NT; ++n) {
            acc[n] = __builtin_amdgcn_wmma_f32_16x16x4_f32(false, xa0, false, Bih[n][0],
                                                           (short)0, acc[n], false, false);
            acc[n] = __builtin_amdgcn_wmma_f32_16x16x4_f32(false, xa1, false, Bih[n][1],
                                                           (short)0, acc[n], false, false);
        }
        // h W_hh^T (K=52, zero-padded beyond 50); n-inner for independent WMMA back-to-back
        #pragma unroll
        for (int c = 0; c < KC; ++c)
            #pragma unroll
            for (int n = 0; n < NT; ++n)
                acc[n] = __builtin_amdgcn_wmma_f32_16x16x4_f32(false, ha[c], false, Bhh[n][c],
                                                               (short)0, acc[n], false, false);

        // tanh + transpose through private LDS (D layout -> A layout); DS in-order per wave
        #pragma unroll
        for (int n = 0; n < NT; ++n)
            #pragma unroll
            for (int r = 0; r < 8; ++r)
                hl[(r + 8 * hi) * RS + 16 * n + lo] = fast_tanh(acc[n][r]);

        #pragma unroll
        for (int c = 0; c < KC; ++c)
            ha[c] = *(const v2f*)&hl[lo * RS + 4 * c + 2 * hi];

        xa0 = nxa0; xa1 = nxa1;
    }

    // fc readout on last hidden state (in LDS): out[b] = fc_b + sum_j h[b][j]*fc_W[j]
    if (lane < 16) {
        float s = fc_b[0];
        #pragma unroll 10
        for (int j = 0; j < H_SZ; ++j)
            s += hl[lo * RS + j] * fc_W[j];
        out[b0 + lo] = s;
    }
}

extern "C" void kernel_launch(void* const* d_in, const int* in_sizes, int n_in,
                              void* d_out, int out_size, void* d_ws, size_t ws_size,
                              hipStream_t stream) {
    (void)in_sizes; (void)n_in; (void)out_size; (void)d_ws; (void)ws_size;
    rnn_scan_kernel<<<dim3(B_SZ / 16), dim3(32), 0, stream>>>(
        (const float*)d_in[0],  // x
        (const float*)d_in[1],  // W_ih
        (const float*)d_in[2],  // W_hh
        (const float*)d_in[3],  // b_ih
        (const float*)d_in[4],  // b_hh
        (const float*)d_in[5],  // fc_W
        (const float*)d_in[6],  // fc_b
        (float*)d_out);
}